// gc_GNN_DropMessage_5841155523231
// MI455X (gfx1250) — compile-verified
//
#include <hip/hip_runtime.h>
#include <hip/hip_bf16.h>

// ---------------------------------------------------------------------------
// GNN (GraphConv + 2x GCN + mean-pool + cosine classifier) for MI455X gfx1250.
// Memory-bound on edge gather/scatter (~2.5 GB -> ~110us at 23.3 TB/s); the
// three N x 128 x 128 GEMMs (~5 GFLOP) use fp32 WMMA (v_wmma_f32_16x16x4_f32)
// to keep full precision since matrix throughput is nowhere near the limiter.
// ---------------------------------------------------------------------------

typedef __attribute__((ext_vector_type(2))) float v2f;
typedef __attribute__((ext_vector_type(8))) float v8f;

#define HDIM 128
#define LDS_STRIDE 132   // 128 + 4 pad: A/B fragment reads hit distinct banks

// ---------------- init: agg = 0, deg = 1 (self-loop) -----------------------
__global__ void init_agg_deg(float* __restrict__ agg, float* __restrict__ deg, int N) {
    int i = blockIdx.x * blockDim.x + threadIdx.x;
    int total = N * HDIM;
    if (i < total) agg[i] = 0.0f;
    if (i < N) deg[i] = 1.0f;
}

// ------- conv1 aggregation: agg[col] += x[row], deg[col] += 1 (wave/edge) ---
__global__ void edge_agg_kernel(const float* __restrict__ x,
                                const int* __restrict__ row,
                                const int* __restrict__ col,
                                float* __restrict__ agg,
                                float* __restrict__ deg, int E) {
    int w = (blockIdx.x * blockDim.x + threadIdx.x) >> 5;
    if (w >= E) return;
    int lane = threadIdx.x & 31;
    int r = row[w], c = col[w];
    if (lane == 0) atomicAdd(&deg[c], 1.0f);
    float4 v = ((const float4*)(x + (size_t)r * HDIM))[lane];
    float* dst = agg + (size_t)c * HDIM + (lane << 2);
    atomicAdd(dst + 0, v.x); atomicAdd(dst + 1, v.y);
    atomicAdd(dst + 2, v.z); atomicAdd(dst + 3, v.w);
}

// ---------------- dinv = rsqrt(deg)  (deg >= 1 due to self loops) -----------
__global__ void rsqrt_kernel(float* __restrict__ deg, int N) {
    int i = blockIdx.x * blockDim.x + threadIdx.x;
    if (i < N) deg[i] = rsqrtf(deg[i]);
}

// ---------------------------------------------------------------------------
// WMMA GEMM: C[N,128] = act(A0)@W0^T (+ A1@W1^T) (+ bias), optional out-relu.
// Block = 256 threads (8 waves, wave32). Block covers 16 rows x 128 cols; wave
// w owns output cols [16w,16w+16). K=128 chained through 32 (or 64) wmma ops.
// Fragment layouts per ISA 7.12.2:
//   A (16x4 f32): lane L: M=L%16, vgpr v holds K = k0 + 2*(L/16) + v
//   B (4x16 f32): lane L: N=L%16, vgpr v holds K = k0 + 2*(L/16) + v
//   C (16x16 f32): lane L: N=L%16, vgpr v holds M = v + 8*(L/16)
// ---------------------------------------------------------------------------
template <bool RELU_A, bool RELU_OUT, bool DUAL, bool BIAS>
__global__ __launch_bounds__(256) void gemm128_wmma(
        const float* __restrict__ A0, const float* __restrict__ A1,
        const float* __restrict__ W0, const float* __restrict__ W1,
        const float* __restrict__ bias, float* __restrict__ Cmat, int N) {
    __shared__ __align__(16) float As[(DUAL ? 2 : 1) * 16 * LDS_STRIDE];

    int tid = threadIdx.x;
    int n0 = blockIdx.x * 16;

    // cooperatively stage A tile(s): 16x128 f32 each, float4 per thread x2
    for (int i = tid; i < 16 * 32; i += 256) {
        int r = i >> 5;
        int c4 = (i & 31) << 2;
        float4 a = make_float4(0.f, 0.f, 0.f, 0.f);
        if (n0 + r < N) a = *(const float4*)(A0 + (size_t)(n0 + r) * HDIM + c4);
        if (RELU_A) {
            a.x = fmaxf(a.x, 0.f); a.y = fmaxf(a.y, 0.f);
            a.z = fmaxf(a.z, 0.f); a.w = fmaxf(a.w, 0.f);
        }
        float* d = &As[r * LDS_STRIDE + c4];
        d[0] = a.x; d[1] = a.y; d[2] = a.z; d[3] = a.w;
    }
    if (DUAL) {
        for (int i = tid; i < 16 * 32; i += 256) {
            int r = i >> 5;
            int c4 = (i & 31) << 2;
            float4 a = make_float4(0.f, 0.f, 0.f, 0.f);
            if (n0 + r < N) a = *(const float4*)(A1 + (size_t)(n0 + r) * HDIM + c4);
            float* d = &As[(16 + r) * LDS_STRIDE + c4];
            d[0] = a.x; d[1] = a.y; d[2] = a.z; d[3] = a.w;
        }
    }
    __syncthreads();

    int wave = tid >> 5;
    int lane = tid & 31;
    int m = lane & 15;
    int khalf = (lane >> 4) << 1;      // 0 or 2
    int ccol = wave * 16 + m;          // this lane's output column

    float bv = 0.0f;
    if (BIAS) bv = bias[ccol];
    v8f acc;
#pragma unroll
    for (int vi = 0; vi < 8; ++vi) acc[vi] = bv;

    const v2f* ar0 = (const v2f*)(As + m * LDS_STRIDE);
    const v2f* wr0 = (const v2f*)(W0 + (size_t)ccol * HDIM);
#pragma unroll
    for (int k0 = 0; k0 < HDIM; k0 += 4) {
        v2f a = ar0[(k0 + khalf) >> 1];
        v2f b = wr0[(k0 + khalf) >> 1];
        acc = __builtin_amdgcn_wmma_f32_16x16x4_f32(false, a, false, b,
                                                    (short)0, acc, false, false);
    }
    if (DUAL) {
        const v2f* ar1 = (const v2f*)(As + (16 + m) * LDS_STRIDE);
        const v2f* wr1 = (const v2f*)(W1 + (size_t)ccol * HDIM);
#pragma unroll
        for (int k0 = 0; k0 < HDIM; k0 += 4) {
            v2f a = ar1[(k0 + khalf) >> 1];
            v2f b = wr1[(k0 + khalf) >> 1];
            acc = __builtin_amdgcn_wmma_f32_16x16x4_f32(false, a, false, b,
                                                        (short)0, acc, false, false);
        }
    }

    int rbase = n0 + ((lane >> 4) << 3);
#pragma unroll
    for (int vi = 0; vi < 8; ++vi) {
        int rr = rbase + vi;
        if (rr < N) {
            float val = acc[vi];
            if (RELU_OUT) val = fmaxf(val, 0.f);
            Cmat[(size_t)rr * HDIM + ccol] = val;
        }
    }
}

// ------- GCN init: out = bias + hp * dinv^2 (self-loop term, no atomics) ----
__global__ void gcn_init_kernel(const float* __restrict__ hp,
                                const float* __restrict__ dinv,
                                const float* __restrict__ bias,
                                float* __restrict__ out, int total) {
    int i = blockIdx.x * blockDim.x + threadIdx.x;
    if (i >= total) return;
    int n = i >> 7;
    int h = i & 127;
    float di = dinv[n];
    out[i] = bias[h] + hp[i] * di * di;
}

// ------- GCN edge scatter: out[col] += hp[row]*dinv[row]*dinv[col] ----------
__global__ void edge_msg_kernel(const float* __restrict__ hp,
                                const int* __restrict__ row,
                                const int* __restrict__ col,
                                const float* __restrict__ dinv,
                                float* __restrict__ out, int E) {
    int w = (blockIdx.x * blockDim.x + threadIdx.x) >> 5;
    if (w >= E) return;
    int lane = threadIdx.x & 31;
    int r = row[w], c = col[w];
    float s = dinv[r] * dinv[c];
    float4 v = ((const float4*)(hp + (size_t)r * HDIM))[lane];
    float* dst = out + (size_t)c * HDIM + (lane << 2);
    atomicAdd(dst + 0, v.x * s); atomicAdd(dst + 1, v.y * s);
    atomicAdd(dst + 2, v.z * s); atomicAdd(dst + 3, v.w * s);
}

// ---------------- pooling ---------------------------------------------------
__global__ void pool_init_kernel(float* __restrict__ gsum, float* __restrict__ gcnt, int G) {
    int i = blockIdx.x * blockDim.x + threadIdx.x;
    if (i < G * HDIM) gsum[i] = 0.0f;
    if (i < G) gcnt[i] = 0.0f;
}

__global__ void pool_kernel(const float* __restrict__ h3, const int* __restrict__ batch,
                            float* __restrict__ gsum, float* __restrict__ gcnt, int N) {
    int w = (blockIdx.x * blockDim.x + threadIdx.x) >> 5;
    if (w >= N) return;
    int lane = threadIdx.x & 31;
    int g = batch[w];
    float4 v = ((const float4*)(h3 + (size_t)w * HDIM))[lane];
    float* dst = gsum + (size_t)g * HDIM + (lane << 2);
    atomicAdd(dst + 0, v.x); atomicAdd(dst + 1, v.y);
    atomicAdd(dst + 2, v.z); atomicAdd(dst + 3, v.w);
    if (lane == 0) atomicAdd(&gcnt[g], 1.0f);
}

// ------- finalize: mean, L2-normalize, cosine classifier --------------------
__global__ __launch_bounds__(128) void finalize_kernel(
        const float* __restrict__ gsum, const float* __restrict__ gcnt,
        const float* __restrict__ Wl, const float* __restrict__ bl,
        float* __restrict__ out_xn, float* __restrict__ out_logits, int C) {
    int g = blockIdx.x;
    int t = threadIdx.x;
    __shared__ float red[HDIM];
    __shared__ float xn_s[HDIM];

    float cnt = fmaxf(gcnt[g], 1.0f);
    float v = gsum[(size_t)g * HDIM + t] / cnt;
    red[t] = v * v;
    __syncthreads();
#pragma unroll
    for (int s = 64; s > 0; s >>= 1) {
        if (t < s) red[t] += red[t + s];
        __syncthreads();
    }
    float nrm = fmaxf(sqrtf(red[0]), 1e-12f);
    float xv = v / nrm;
    xn_s[t] = xv;
    out_xn[(size_t)g * HDIM + t] = xv;
    __syncthreads();
    if (t < C) {
        float dot = 0.f, w2 = 0.f;
        for (int h = 0; h < HDIM; ++h) {
            float w = Wl[(size_t)t * HDIM + h];
            dot += xn_s[h] * w;
            w2 += w * w;
        }
        float wn = fmaxf(sqrtf(w2), 1e-12f);
        out_logits[(size_t)g * C + t] = dot / wn + bl[t];
    }
}

// ---------------------------------------------------------------------------
extern "C" void kernel_launch(void* const* d_in, const int* in_sizes, int n_in,
                              void* d_out, int out_size, void* d_ws, size_t ws_size,
                              hipStream_t stream) {
    const float* x       = (const float*)d_in[0];
    const int*   ei      = (const int*)d_in[1];
    const int*   batch   = (const int*)d_in[2];
    const float* W1_rel  = (const float*)d_in[3];
    const float* b1      = (const float*)d_in[4];
    const float* W1_root = (const float*)d_in[5];
    const float* W2      = (const float*)d_in[6];
    const float* b2      = (const float*)d_in[7];
    const float* W3      = (const float*)d_in[8];
    const float* b3      = (const float*)d_in[9];
    const float* Wl      = (const float*)d_in[10];
    const float* bl      = (const float*)d_in[11];

    int N = in_sizes[0] / HDIM;
    int E = in_sizes[1] / 2;
    int C = in_sizes[10] / HDIM;
    int G = out_size / (HDIM + C);
    const int* row = ei;
    const int* col = ei + E;

    float* ws = (float*)d_ws;
    size_t NH = (size_t)N * HDIM;
    float* bufA = ws;            // agg -> hp(conv2) -> h3
    float* bufB = ws + NH;       // h1  -> hp(conv3)
    float* bufC = ws + 2 * NH;   // gcn accumulation (conv2 pre-relu)
    float* deg  = ws + 3 * NH;   // deg, then dinv in place
    float* gsum = deg + N;
    float* gcnt = gsum + (size_t)G * HDIM;

    float* out_xn     = (float*)d_out;
    float* out_logits = (float*)d_out + (size_t)G * HDIM;

    int totNH = N * HDIM;
    unsigned bNH = (unsigned)((totNH + 255) / 256);
    unsigned bE  = (unsigned)((E + 7) / 8);   // 8 waves per 256-thread block
    unsigned bN  = (unsigned)((N + 7) / 8);
    unsigned bG  = (unsigned)((N + 15) / 16); // gemm row tiles

    // conv1
    init_agg_deg<<<bNH, 256, 0, stream>>>(bufA, deg, N);
    edge_agg_kernel<<<bE, 256, 0, stream>>>(x, row, col, bufA, deg, E);
    rsqrt_kernel<<<(unsigned)((N + 255) / 256), 256, 0, stream>>>(deg, N);
    gemm128_wmma<false, true, true, true><<<bG, 256, 0, stream>>>(
        bufA, x, W1_rel, W1_root, b1, bufB, N);            // h1 = relu(...)

    // conv2 (GCN)
    gemm128_wmma<false, false, false, false><<<bG, 256, 0, stream>>>(
        bufB, nullptr, W2, nullptr, nullptr, bufA, N);     // hp = h1 @ W2^T
    gcn_init_kernel<<<bNH, 256, 0, stream>>>(bufA, deg, b2, bufC, totNH);
    edge_msg_kernel<<<bE, 256, 0, stream>>>(bufA, row, col, deg, bufC, E);

    // conv3 (GCN; relu of conv2 folded into A-load)
    gemm128_wmma<true, false, false, false><<<bG, 256, 0, stream>>>(
        bufC, nullptr, W3, nullptr, nullptr, bufB, N);     // hp2 = relu(h2) @ W3^T
    gcn_init_kernel<<<bNH, 256, 0, stream>>>(bufB, deg, b3, bufA, totNH);
    edge_msg_kernel<<<bE, 256, 0, stream>>>(bufB, row, col, deg, bufA, E);

    // pool + classifier
    pool_init_kernel<<<(unsigned)((G * HDIM + 255) / 256), 256, 0, stream>>>(gsum, gcnt, G);
    pool_kernel<<<bN, 256, 0, stream>>>(bufA, batch, gsum, gcnt, N);
    finalize_kernel<<<(unsigned)G, 128, 0, stream>>>(gsum, gcnt, Wl, bl,
                                                     out_xn, out_logits, C);
}